// MaxAggregator_10385230921951
// MI455X (gfx1250) — compile-verified
//
#include <hip/hip_runtime.h>
#include <stdint.h>

#define NN 100000
#define NE 1600000
#define D 64
#define LDSROW 68   // padded LDS row pitch (floats) -> bank-conflict-free strided reads

typedef __attribute__((ext_vector_type(2))) float v2f;
typedef __attribute__((ext_vector_type(8))) float v8f;

// Monotone order-preserving float<->uint encoding so we can use hardware
// GLOBAL_ATOMIC_MAX_U32 (deterministic; max is commutative/associative).
__device__ __forceinline__ unsigned int enc_f(float f) {
  unsigned int u = __float_as_uint(f);
  return (u & 0x80000000u) ? ~u : (u | 0x80000000u);
}
__device__ __forceinline__ float dec_f(unsigned int u) {
  return __uint_as_float((u & 0x80000000u) ? (u & 0x7FFFFFFFu) : ~u);
}

// ---------------------------------------------------------------------------
// 1) init workspace: encoded-max array to 0 (== -infinity), counts to 0
// ---------------------------------------------------------------------------
__global__ void init_ws_kernel(unsigned int* __restrict__ enc,
                               unsigned int* __restrict__ cnt) {
  int t = blockIdx.x * blockDim.x + threadIdx.x;
  if (t < NN * D) enc[t] = 0u;
  if (t < NN)     cnt[t] = 0u;
}

// ---------------------------------------------------------------------------
// 2) scatter-max: 16 threads per edge, float4 per thread; L2-resident atomics
// ---------------------------------------------------------------------------
__global__ void scatter_max_kernel(const float* __restrict__ x,
                                   const int* __restrict__ row,
                                   const int* __restrict__ col,
                                   unsigned int* __restrict__ enc,
                                   unsigned int* __restrict__ cnt) {
  int t = blockIdx.x * blockDim.x + threadIdx.x;   // [0, NE*16) fits in int32
  if (t >= NE * 16) return;
  int e  = t >> 4;
  int k4 = (t & 15) << 2;
  int r = row[e];
  int c = col[e];
  const float4 v = *(const float4*)(x + (size_t)c * D + k4);
  unsigned int* dst = enc + (size_t)r * D + k4;
  atomicMax(dst + 0, enc_f(v.x));
  atomicMax(dst + 1, enc_f(v.y));
  atomicMax(dst + 2, enc_f(v.z));
  atomicMax(dst + 3, enc_f(v.w));
  if ((t & 15) == 0) atomicAdd(cnt + r, 1u);
}

// ---------------------------------------------------------------------------
// 3) decode + isolated-node fallback + (agg @ W^T + b) via V_WMMA_F32_16X16X4_F32
//    block = 256 threads = 8 waves; wave -> 16x64 output tile
// ---------------------------------------------------------------------------
__launch_bounds__(256)
__global__ void agg_linear_kernel(const float* __restrict__ x,
                                  const float* __restrict__ W,
                                  const float* __restrict__ bias,
                                  const unsigned int* __restrict__ enc,
                                  const unsigned int* __restrict__ cnt,
                                  float* __restrict__ out) {
  __shared__ float sA[128 * LDSROW];  // 34.8 KB
  __shared__ float sW[64 * LDSROW];   // 17.4 KB  (WGP LDS pool = 320 KB)
  const int tid = threadIdx.x;
  const int nodeBase = blockIdx.x * 128;

  // Stage W [out=64][in=64] row-major into padded LDS (coalesced float4 loads).
  for (int i = tid; i < 64 * 16; i += 256) {
    int j  = i >> 4;
    int k4 = (i & 15) << 2;
    const float4 w4 = *(const float4*)(W + j * 64 + k4);
    float* d = sW + j * LDSROW + k4;
    d[0] = w4.x; d[1] = w4.y; d[2] = w4.z; d[3] = w4.w;
  }

  // Stage aggregated features: decode encoded max, fall back to x for cnt==0.
  for (int i = tid; i < 128 * 16; i += 256) {
    int nLoc = i >> 4;
    int node = nodeBase + nLoc;
    int k4   = (i & 15) << 2;
    float4 v = make_float4(0.f, 0.f, 0.f, 0.f);
    if (node < NN) {
      if (cnt[node] > 0u) {
        const uint4 e4 = *(const uint4*)(enc + (size_t)node * D + k4);
        v.x = dec_f(e4.x); v.y = dec_f(e4.y); v.z = dec_f(e4.z); v.w = dec_f(e4.w);
      } else {
        v = *(const float4*)(x + (size_t)node * D + k4);
      }
    }
    float* d = sA + nLoc * LDSROW + k4;
    d[0] = v.x; d[1] = v.y; d[2] = v.z; d[3] = v.w;
  }
  __syncthreads();

  const int wave = tid >> 5;          // wave32
  const int lane = tid & 31;
  const int tileNode = nodeBase + wave * 16;
  if (tileNode >= NN) return;         // wave-uniform: EXEC stays all-ones for WMMA

  const int m    = lane & 15;         // row (A) / column (B,C,D) within tile
  const int half = lane >> 4;         // K pair select for A/B fragments

  v8f acc0 = {}; v8f acc1 = {}; v8f acc2 = {}; v8f acc3 = {};
  const float* aRow = sA + (wave * 16 + m) * LDSROW + 2 * half;
  const float* bRow = sW + m * LDSROW + 2 * half;

  #pragma unroll
  for (int s = 0; s < 16; ++s) {
    const int k0 = s * 4;
    v2f a;  a.x  = aRow[k0];                    a.y  = aRow[k0 + 1];
    v2f b0; b0.x = bRow[k0];                    b0.y = bRow[k0 + 1];
    v2f b1; b1.x = bRow[k0 + 16 * LDSROW];      b1.y = bRow[k0 + 16 * LDSROW + 1];
    v2f b2; b2.x = bRow[k0 + 32 * LDSROW];      b2.y = bRow[k0 + 32 * LDSROW + 1];
    v2f b3; b3.x = bRow[k0 + 48 * LDSROW];      b3.y = bRow[k0 + 48 * LDSROW + 1];
    acc0 = __builtin_amdgcn_wmma_f32_16x16x4_f32(false, a, false, b0, (short)0, acc0, false, false);
    acc1 = __builtin_amdgcn_wmma_f32_16x16x4_f32(false, a, false, b1, (short)0, acc1, false, false);
    acc2 = __builtin_amdgcn_wmma_f32_16x16x4_f32(false, a, false, b2, (short)0, acc2, false, false);
    acc3 = __builtin_amdgcn_wmma_f32_16x16x4_f32(false, a, false, b3, (short)0, acc3, false, false);
  }

  // D layout: VGPR r -> M = r + 8*half ; N = lane&15 within each 16-col tile.
  const int mBase = half * 8;
  float* outRow = out + (size_t)(tileNode + mBase) * D + m;
  {
    const float bj = bias[m];
    #pragma unroll
    for (int r = 0; r < 8; ++r) outRow[(size_t)r * D] = acc0[r] + bj;
  }
  {
    const float bj = bias[16 + m];
    #pragma unroll
    for (int r = 0; r < 8; ++r) outRow[(size_t)r * D + 16] = acc1[r] + bj;
  }
  {
    const float bj = bias[32 + m];
    #pragma unroll
    for (int r = 0; r < 8; ++r) outRow[(size_t)r * D + 32] = acc2[r] + bj;
  }
  {
    const float bj = bias[48 + m];
    #pragma unroll
    for (int r = 0; r < 8; ++r) outRow[(size_t)r * D + 48] = acc3[r] + bj;
  }
}

// ---------------------------------------------------------------------------
extern "C" void kernel_launch(void* const* d_in, const int* in_sizes, int n_in,
                              void* d_out, int out_size, void* d_ws, size_t ws_size,
                              hipStream_t stream) {
  const float* x    = (const float*)d_in[0];   // [NN, 64]
  const float* W    = (const float*)d_in[1];   // [64, 64] (out, in)
  const float* b    = (const float*)d_in[2];   // [64]
  const int*   edge = (const int*)d_in[3];     // [2, NE]
  const int*   row  = edge;
  const int*   col  = edge + NE;
  float* out = (float*)d_out;

  unsigned int* enc = (unsigned int*)d_ws;            // NN*64 u32 (25.6 MB)
  unsigned int* cnt = enc + (size_t)NN * D;           // NN u32   (0.4 MB)

  {
    int blocks = (NN * D + 255) / 256;                // covers both enc and cnt
    init_ws_kernel<<<blocks, 256, 0, stream>>>(enc, cnt);
  }
  {
    int total  = NE * 16;
    int blocks = (total + 255) / 256;
    scatter_max_kernel<<<blocks, 256, 0, stream>>>(x, row, col, enc, cnt);
  }
  {
    int blocks = (NN + 127) / 128;
    agg_linear_kernel<<<blocks, 256, 0, stream>>>(x, W, b, enc, cnt, out);
  }
}